// DeformConv2d_274877907552
// MI455X (gfx1250) — compile-verified
//
#include <hip/hip_runtime.h>
#include <hip/hip_bf16.h>

// Problem constants (b=4, c=64, h=w=128, ks=3)
#define BB    4
#define CC    64
#define HH    128
#define WW    128
#define HW    (HH * WW)          // 16384
#define NPOS  9
#define OFFC  (2 * NPOS)         // 18 offset channels
#define KTOT  (CC * NPOS)        // 576 = GEMM K
#define TILE  32                 // spatial positions per block (N-tile)
#define LSTR  40                 // LDS row stride in floats (32 + 8 pad -> conflict-free B frag)

typedef __attribute__((ext_vector_type(2))) float v2f;
typedef __attribute__((ext_vector_type(8))) float v8f;

// ---------------------------------------------------------------------------
// Kernel 1: offset = conv3x3(x1, p_w) + p_b   -> (4, 18, 128, 128) workspace
// ---------------------------------------------------------------------------
__global__ __launch_bounds__(256) void offset_conv_kernel(
    const float* __restrict__ x1, const float* __restrict__ pw,
    const float* __restrict__ pb, float* __restrict__ off) {
  int idx = blockIdx.x * 256 + threadIdx.x;
  if (idx >= BB * OFFC * HW) return;
  int x  = idx & (WW - 1);
  int y  = (idx >> 7) & (HH - 1);
  int oc = (idx >> 14) % OFFC;
  int b  = idx / (OFFC * HW);

  float acc = pb[oc];
  const float* xin  = x1 + (size_t)b * CC * HW;
  const float* wrow = pw + oc * CC * 9;
  for (int ci = 0; ci < CC; ++ci) {
    const float* xc = xin + ci * HW;
    const float* wc = wrow + ci * 9;
#pragma unroll
    for (int ky = 0; ky < 3; ++ky) {
      int iy = y + ky - 1;
      if (iy < 0 || iy >= HH) continue;
#pragma unroll
      for (int kx = 0; kx < 3; ++kx) {
        int ix = x + kx - 1;
        if (ix < 0 || ix >= WW) continue;
        acc = fmaf(xc[iy * WW + ix], wc[ky * 3 + kx], acc);
      }
    }
  }
  off[idx] = acc;
}

// ---------------------------------------------------------------------------
// Kernel 2 (fused): bilinear sampling -> LDS B-tile (K=576 x N=32),
// then fp32 WMMA GEMM  out[co, pos] = conv_w[co, k] * B[k, pos].
// 256 threads = 8 waves; wave (cotile, nsub) owns one 16x16 C tile.
// ---------------------------------------------------------------------------
__global__ __launch_bounds__(256) void deform_wmma_kernel(
    const float* __restrict__ x0, const float* __restrict__ off,
    const float* __restrict__ Wg, float* __restrict__ out) {
  __shared__ float sB[KTOT * LSTR];        // 576*40*4 = 92160 B

  const int tid   = threadIdx.x;
  const int sbase = blockIdx.x * TILE;     // flat (b,y,x) position base
  const int b     = sbase >> 14;           // 16384 positions per batch image
  const int rem   = sbase & (HW - 1);
  const int y     = rem >> 7;
  const int xb    = rem & (WW - 1);        // tile spans x = xb .. xb+31 (same row)

  const int wave   = tid >> 5;
  const int lane   = tid & 31;
  const int cotile = wave >> 1;            // 0..3  -> co base
  const int nsub   = wave & 1;             // 0..1  -> position sub-tile
  const int m      = lane & 15;
  const int kk     = lane >> 4;            // half-wave selector

  // Prefetch this wave's A rows (weights) so they overlap the sampling phase.
  {
    int co = cotile * 16 + m;
    __builtin_prefetch(Wg + (size_t)co * KTOT + kk * (KTOT / 2), 0, 3);
  }

  // ---- Phase 1: bilinear sampling into LDS (K x N layout) ----
  const float* xb0 = x0 + (size_t)b * CC * HW;
  for (int pair = tid; pair < TILE * NPOS; pair += 256) {
    int sl = pair / NPOS;                  // local position 0..31
    int n  = pair - sl * NPOS;             // kernel tap 0..8
    int ki = n / 3, kj = n - ki * 3;
    int x  = xb + sl;

    float ox = off[((b * OFFC + n) * HH + y) * WW + x];
    float oy = off[((b * OFFC + NPOS + n) * HH + y) * WW + x];
    // p = p0 + p_n + offset  (p0 is 1-indexed; p_n = ki-1 / kj-1)
    float pxr = (float)(y + ki) + ox;
    float pyr = (float)(x + kj) + oy;

    float flx = floorf(pxr), fly = floorf(pyr);
    int iltx = min(max((int)flx, 0), HH + 1);       // padded coords, Hp-1 = 129
    int ilty = min(max((int)fly, 0), WW + 1);
    int irbx = min(max((int)flx + 1, 0), HH + 1);
    int irby = min(max((int)fly + 1, 0), WW + 1);
    float px = fminf(fmaxf(pxr, 0.f), (float)(HH + 1));
    float py = fminf(fmaxf(pyr, 0.f), (float)(WW + 1));

    float fxl = (float)iltx, fyl = (float)ilty;
    float fxr = (float)irbx, fyr = (float)irby;
    float glt = (1.f + (fxl - px)) * (1.f + (fyl - py));
    float grb = (1.f - (fxr - px)) * (1.f - (fyr - py));
    float glb = (1.f + (fxl - px)) * (1.f - (fyr - py));
    float grt = (1.f - (fxr - px)) * (1.f + (fyl - py));

    // Fold padded-image zero-fill into the weights (branchless inner loop):
    // a tap landing in the 1-wide padding ring contributes 0, so zero its
    // weight and clamp its gather index into the real image.
    bool vxl = (iltx >= 1) && (iltx <= HH), vxr = (irbx >= 1) && (irbx <= HH);
    bool vyl = (ilty >= 1) && (ilty <= WW), vyr = (irby >= 1) && (irby <= WW);
    if (!(vxl && vyl)) glt = 0.f;
    if (!(vxr && vyr)) grb = 0.f;
    if (!(vxl && vyr)) glb = 0.f;
    if (!(vxr && vyl)) grt = 0.f;
    int il = min(max(iltx - 1, 0), HH - 1);   // unpadded row/col, always valid
    int ir = min(max(irbx - 1, 0), HH - 1);
    int jl = min(max(ilty - 1, 0), WW - 1);
    int jr = min(max(irby - 1, 0), WW - 1);
    int o_lt = il * WW + jl;
    int o_rb = ir * WW + jr;
    int o_lb = il * WW + jr;
    int o_rt = ir * WW + jl;

    float* dst = &sB[n * LSTR + sl];       // row k = ci*9 + n
    const float* src = xb0;
#pragma unroll 4
    for (int ci = 0; ci < CC; ++ci) {
      float v = glt * src[o_lt];
      v = fmaf(grb, src[o_rb], v);
      v = fmaf(glb, src[o_lb], v);
      v = fmaf(grt, src[o_rt], v);
      dst[0] = v;
      dst += NPOS * LSTR;
      src += HW;
    }
  }
  __syncthreads();

  // ---- Phase 2: D(16x16) += A(16x4) x B(4x16) over K=576 (fp32 WMMA) ----
  // A 16x4 layout: lanes 0-15 VGPR{0,1}=K{0,1}; lanes 16-31 VGPR{0,1}=K{2,3}
  const int co   = cotile * 16 + m;
  const int np   = nsub * 16 + m;          // B/C column (N index)
  const int krow = kk * 2;                 // half-wave K rows for A and B
  const float* arow = Wg + (size_t)co * KTOT + krow;

  v8f acc = {};
#pragma unroll 4
  for (int kb = 0; kb < KTOT; kb += 4) {
    v2f a = *reinterpret_cast<const v2f*>(arow + kb);      // W[co][kb+krow+{0,1}]
    v2f bv;
    bv.x = sB[(kb + krow    ) * LSTR + np];
    bv.y = sB[(kb + krow + 1) * LSTR + np];
    acc = __builtin_amdgcn_wmma_f32_16x16x4_f32(
        /*neg_a=*/false, a, /*neg_b=*/false, bv,
        /*c_mod=*/(short)0, acc, /*reuse_a=*/false, /*reuse_b=*/false);
  }

  // ---- Write C: VGPR j -> M = j + 8*kk, N = lane&15 (+16*nsub) ----
  float* obase = out + (size_t)b * CC * HW + rem + nsub * 16 + m;
#pragma unroll
  for (int j = 0; j < 8; ++j) {
    int mo = cotile * 16 + j + 8 * kk;
    obase[(size_t)mo * HW] = acc[j];
  }
}

// ---------------------------------------------------------------------------
extern "C" void kernel_launch(void* const* d_in, const int* in_sizes, int n_in,
                              void* d_out, int out_size, void* d_ws, size_t ws_size,
                              hipStream_t stream) {
  const float* x0     = (const float*)d_in[0];
  const float* x1     = (const float*)d_in[1];
  const float* p_w    = (const float*)d_in[2];
  const float* p_b    = (const float*)d_in[3];
  const float* conv_w = (const float*)d_in[4];
  float* out = (float*)d_out;
  float* off = (float*)d_ws;               // 4*18*128*128 floats = 4.72 MB

  // Kernel 1: offset field
  offset_conv_kernel<<<(BB * OFFC * HW + 255) / 256, 256, 0, stream>>>(
      x1, p_w, p_b, off);

  // Kernel 2: fused sampling + fp32 WMMA GEMM
  const int npos_total = BB * HW;          // 65536 positions
  deform_wmma_kernel<<<npos_total / TILE, 256, 0, stream>>>(
      x0, off, conv_w, out);
}